// Point_Embedding_85323820302522
// MI455X (gfx1250) — compile-verified
//
#include <hip/hip_runtime.h>

typedef __attribute__((ext_vector_type(2))) float v2f;
typedef __attribute__((ext_vector_type(8))) float v8f;

#define BATCH 8
#define NPTS 4096
#define KNN 20
#define C1IN 6
#define C2IN 128
#define NEG_SLOPE 0.2f
#define BN_EPS 1e-5f

__device__ __forceinline__ v8f wmma_f32_16x16x4(v2f a, v2f b, v8f c) {
  // D = A(16x4 f32) * B(4x16 f32) + C(16x16 f32)
  return __builtin_amdgcn_wmma_f32_16x16x4_f32(false, a, false, b, (short)0, c,
                                               false, false);
}

// ---------------------------------------------------------------------------
// Stage 1: brute-force kNN of a (B,N,3) against b (B,N,3), k=20
// ---------------------------------------------------------------------------
__global__ __launch_bounds__(128) void knn1_kernel(const float* __restrict__ A,
                                                   const float* __restrict__ Bp,
                                                   int* __restrict__ idx) {
  const int batch = blockIdx.y;
  const int q = blockIdx.x * 128 + threadIdx.x;   // grid.x = NPTS/128 -> always valid
  const float* ab = A + (size_t)batch * NPTS * 3;
  const float* bb = Bp + (size_t)batch * NPTS * 3;
  __shared__ float sref[1024 * 3];
  __shared__ float sd[128 * KNN];
  __shared__ int si[128 * KNN];
  float qx = ab[q * 3 + 0], qy = ab[q * 3 + 1], qz = ab[q * 3 + 2];
  float* md = &sd[threadIdx.x * KNN];
  int* mi = &si[threadIdx.x * KNN];
#pragma unroll
  for (int j = 0; j < KNN; ++j) { md[j] = 3.0e38f; mi[j] = 0; }
  float wmax = 3.0e38f;
  int wslot = 0;
  for (int tile = 0; tile < NPTS; tile += 1024) {
    __syncthreads();
    for (int t = threadIdx.x; t < 1024 * 3; t += 128) sref[t] = bb[(size_t)tile * 3 + t];
    __syncthreads();
    for (int m = 0; m < 1024; ++m) {
      float dx = qx - sref[m * 3 + 0];
      float dy = qy - sref[m * 3 + 1];
      float dz = qz - sref[m * 3 + 2];
      float d = dx * dx + dy * dy + dz * dz;
      if (d < wmax) {
        md[wslot] = d;
        mi[wslot] = tile + m;
        wmax = md[0];
        wslot = 0;
#pragma unroll
        for (int j = 1; j < KNN; ++j) {
          float v = md[j];
          if (v > wmax) { wmax = v; wslot = j; }
        }
      }
    }
  }
  int* out = idx + ((size_t)batch * NPTS + q) * KNN;
#pragma unroll
  for (int j = 0; j < KNN; ++j) out[j] = mi[j];
}

// ---------------------------------------------------------------------------
// Stage 2: EdgeConv1 GEMM (80 x 8pad x 64 per block) via WMMA f32 16x16x4.
// W1 staged zero-padded in LDS (uniform, no divergent global loads).
// ---------------------------------------------------------------------------
__global__ __launch_bounds__(64) void edgeconv1_kernel(
    const float* __restrict__ A, const int* __restrict__ idx,
    const float* __restrict__ W1, float* __restrict__ hmax,
    float* __restrict__ hmin, float* __restrict__ sums,
    float* __restrict__ sumsq) {
  const int batch = blockIdx.y;
  const int nbase = blockIdx.x * 4;   // 4 points -> 80 feat rows (5 m-tiles exactly)
  __shared__ float feat[80][8];       // K padded 6 -> 8
  __shared__ float Wpad[8][64];       // B-layout: Wpad[k][o], zeros for k>=6
  __shared__ float h[80][64];
  const float* ab = A + (size_t)batch * NPTS * 3;
  {
    int o = threadIdx.x;  // 64 threads == 64 out channels
#pragma unroll
    for (int k = 0; k < 8; ++k) Wpad[k][o] = (k < C1IN) ? W1[o * C1IN + k] : 0.f;
  }
  for (int r = threadIdx.x; r < 80; r += 64) {
    int p = r / KNN, kk = r % KNN;
    int n = nbase + p;
    int nb = idx[((size_t)batch * NPTS + n) * KNN + kk];
    float cx = ab[n * 3 + 0], cy = ab[n * 3 + 1], cz = ab[n * 3 + 2];
    float nx = ab[(size_t)nb * 3 + 0], ny = ab[(size_t)nb * 3 + 1], nz = ab[(size_t)nb * 3 + 2];
    feat[r][0] = nx - cx; feat[r][1] = ny - cy; feat[r][2] = nz - cz;
    feat[r][3] = cx; feat[r][4] = cy; feat[r][5] = cz;
    feat[r][6] = 0.f; feat[r][7] = 0.f;
  }
  __syncthreads();
  const int wave = threadIdx.x >> 5;  // 2 waves
  const int lane = threadIdx.x & 31;
  const int l16 = lane & 15;
  const int hi = lane >> 4;           // fragments: lanes>=16 hold K+2,K+3
  for (int nt = wave * 2; nt < wave * 2 + 2; ++nt) {   // 4 n-tiles of 16 channels
    v2f bfrag[2];
#pragma unroll
    for (int kc = 0; kc < 2; ++kc) {
      int k0 = kc * 4 + hi * 2;
      bfrag[kc].x = Wpad[k0][nt * 16 + l16];
      bfrag[kc].y = Wpad[k0 + 1][nt * 16 + l16];
    }
    for (int mt = 0; mt < 5; ++mt) {
      v8f acc = {};
#pragma unroll
      for (int kc = 0; kc < 2; ++kc) {
        int k0 = kc * 4 + hi * 2;
        v2f av;
        av.x = feat[mt * 16 + l16][k0];
        av.y = feat[mt * 16 + l16][k0 + 1];
        acc = wmma_f32_16x16x4(av, bfrag[kc], acc);
      }
#pragma unroll
      for (int v = 0; v < 8; ++v) h[mt * 16 + v + hi * 8][nt * 16 + l16] = acc[v];
    }
  }
  __syncthreads();
  int o = threadIdx.x;  // 64 threads == 64 channels
  float s = 0.f, s2 = 0.f;
  for (int p = 0; p < 4; ++p) {
    float mx = -3.0e38f, mn = 3.0e38f;
    for (int kk = 0; kk < KNN; ++kk) {
      float v = h[p * KNN + kk][o];
      mx = fmaxf(mx, v);
      mn = fminf(mn, v);
      s += v;
      s2 += v * v;
    }
    size_t bn = (size_t)batch * NPTS + nbase + p;
    hmax[bn * 64 + o] = mx;
    hmin[bn * 64 + o] = mn;
  }
  atomicAdd(&sums[o], s);
  atomicAdd(&sumsq[o], s2);
}

// ---------------------------------------------------------------------------
// BN finalize: scale/shift per channel
// ---------------------------------------------------------------------------
__global__ __launch_bounds__(64) void bn_finalize_kernel(
    const float* __restrict__ sums, const float* __restrict__ sumsq,
    const float* __restrict__ gamma, const float* __restrict__ beta,
    float* __restrict__ scale, float* __restrict__ shift, float inv_cnt) {
  int o = threadIdx.x;
  float mean = sums[o] * inv_cnt;
  float var = sumsq[o] * inv_cnt - mean * mean;
  float sc = gamma[o] * rsqrtf(var + BN_EPS);
  scale[o] = sc;
  shift[o] = beta[o] - mean * sc;
}

// ---------------------------------------------------------------------------
// Apply BN+LeakyReLU to max/min (monotonicity trick), produce x1 and |x1|^2
// ---------------------------------------------------------------------------
__global__ __launch_bounds__(256) void apply1_kernel(
    const float* __restrict__ hmax, const float* __restrict__ hmin,
    const float* __restrict__ scale, const float* __restrict__ shift,
    float* __restrict__ x1, float* __restrict__ xnorm2) {
  size_t bn = (size_t)blockIdx.x * 256 + threadIdx.x;
  float nrm = 0.f;
  for (int o = 0; o < 64; ++o) {
    float sc = scale[o];
    float hv = (sc >= 0.f) ? hmax[bn * 64 + o] : hmin[bn * 64 + o];
    float v = sc * hv + shift[o];
    v = (v >= 0.f) ? v : NEG_SLOPE * v;
    x1[bn * 64 + o] = v;
    nrm += v * v;
  }
  xnorm2[bn] = nrm;
}

// ---------------------------------------------------------------------------
// Stage 3: kNN of x1 vs x1 (C=64) via WMMA Gram tiles + streaming top-20.
// 128 threads = 4 waves; each wave owns one 16-row query m-tile.
// A-fragments (query tile) preloaded to registers once; inner loop is
// 2 ds-loads + 1 wmma per k-step.
// ---------------------------------------------------------------------------
__global__ __launch_bounds__(128) void knn2_kernel(const float* __restrict__ x1,
                                                   const float* __restrict__ xnorm2,
                                                   int* __restrict__ idx) {
  const int batch = blockIdx.y;
  const int qbase = blockIdx.x * 64;
  __shared__ float Xq[64][64];
  __shared__ float Xr[16][64];
  __shared__ float qn2[64];
  __shared__ float rn2[16];
  __shared__ float Dt[4][16][16];
  __shared__ float topd[64 * KNN];
  __shared__ int topi[64 * KNN];
  const float* xb = x1 + (size_t)batch * NPTS * 64;
  const float* nb2 = xnorm2 + (size_t)batch * NPTS;
  for (int t = threadIdx.x; t < 64 * 64; t += 128)
    Xq[t >> 6][t & 63] = xb[(size_t)(qbase + (t >> 6)) * 64 + (t & 63)];
  if (threadIdx.x < 64) qn2[threadIdx.x] = nb2[qbase + threadIdx.x];
  for (int t = threadIdx.x; t < 64 * KNN; t += 128) { topd[t] = 3.0e38f; topi[t] = 0; }
  const int wave = threadIdx.x >> 5;
  const int lane = threadIdx.x & 31;
  const int l16 = lane & 15;
  const int hi = lane >> 4;
  __syncthreads();
  v2f afrag[16];  // invariant across all ref tiles
#pragma unroll
  for (int kc = 0; kc < 16; ++kc) {
    int k0 = kc * 4 + hi * 2;
    afrag[kc].x = Xq[wave * 16 + l16][k0];
    afrag[kc].y = Xq[wave * 16 + l16][k0 + 1];
  }
  float wmax = 3.0e38f;
  int wslot = 0;
  for (int rt = 0; rt < NPTS / 16; ++rt) {
    __syncthreads();
    for (int t = threadIdx.x; t < 16 * 64; t += 128)
      Xr[t >> 6][t & 63] = xb[(size_t)(rt * 16 + (t >> 6)) * 64 + (t & 63)];
    if (threadIdx.x < 16) rn2[threadIdx.x] = nb2[rt * 16 + threadIdx.x];
    __syncthreads();
    v8f acc = {};   // S = Xq_tile (16x64) * Xr_tile^T (64x16), 16 WMMA steps
#pragma unroll
    for (int kc = 0; kc < 16; ++kc) {
      int k0 = kc * 4 + hi * 2;
      v2f bv;
      bv.x = Xr[l16][k0];
      bv.y = Xr[l16][k0 + 1];
      acc = wmma_f32_16x16x4(afrag[kc], bv, acc);
    }
#pragma unroll
    for (int v = 0; v < 8; ++v) {
      int m = v + hi * 8;
      Dt[wave][m][l16] = qn2[wave * 16 + m] - 2.f * acc[v] + rn2[l16];
    }
    __syncthreads();
    if (lane < 16) {   // lanes 0..15 each own one query row of this wave's m-tile
      int row = wave * 16 + lane;
      float* md = &topd[row * KNN];
      int* mi = &topi[row * KNN];
      for (int c = 0; c < 16; ++c) {
        float d = Dt[wave][lane][c];
        if (d < wmax) {
          md[wslot] = d;
          mi[wslot] = rt * 16 + c;
          wmax = md[0];
          wslot = 0;
#pragma unroll
          for (int j = 1; j < KNN; ++j) {
            float v = md[j];
            if (v > wmax) { wmax = v; wslot = j; }
          }
        }
      }
    }
  }
  __syncthreads();
  if (lane < 16) {
    int row = wave * 16 + lane;
    int* out = idx + ((size_t)batch * NPTS + qbase + row) * KNN;
#pragma unroll
    for (int j = 0; j < KNN; ++j) out[j] = topi[row * KNN + j];
  }
}

// ---------------------------------------------------------------------------
// Stage 4: EdgeConv2 GEMM (80 x 128 x 64 per block) via WMMA.
// k-chunk outer / m-tile inner: each W2 B-fragment loaded once, reused 5x.
// ---------------------------------------------------------------------------
__global__ __launch_bounds__(128) void edgeconv2_kernel(
    const float* __restrict__ x1, const int* __restrict__ idx,
    const float* __restrict__ W2, float* __restrict__ hmax,
    float* __restrict__ hmin, float* __restrict__ sums,
    float* __restrict__ sumsq) {
  const int batch = blockIdx.y;
  const int nbase = blockIdx.x * 4;
  __shared__ float feat[80][128];
  __shared__ float h[80][64];
  const float* xb = x1 + (size_t)batch * NPTS * 64;
  const int c = threadIdx.x;  // 0..127 feature column
  for (int r = 0; r < 80; ++r) {
    int p = r / KNN, kk = r % KNN;
    int n = nbase + p;
    int nbi = idx[((size_t)batch * NPTS + n) * KNN + kk];
    float v;
    if (c < 64)
      v = xb[(size_t)nbi * 64 + c] - xb[(size_t)n * 64 + c];
    else
      v = xb[(size_t)n * 64 + (c - 64)];
    feat[r][c] = v;
  }
  __syncthreads();
  const int wave = threadIdx.x >> 5;  // wave == n-tile (4 waves, 4 n-tiles)
  const int lane = threadIdx.x & 31;
  const int l16 = lane & 15;
  const int hi = lane >> 4;
  v8f acc[5] = {};
  for (int kc = 0; kc < 32; ++kc) {  // K = 128
    int k0 = kc * 4 + hi * 2;
    v2f bv;
    int o = wave * 16 + l16;
    bv.x = W2[o * C2IN + k0];
    bv.y = W2[o * C2IN + k0 + 1];
#pragma unroll
    for (int mt = 0; mt < 5; ++mt) {
      v2f av;
      av.x = feat[mt * 16 + l16][k0];
      av.y = feat[mt * 16 + l16][k0 + 1];
      acc[mt] = wmma_f32_16x16x4(av, bv, acc[mt]);
    }
  }
#pragma unroll
  for (int mt = 0; mt < 5; ++mt)
#pragma unroll
    for (int v = 0; v < 8; ++v)
      h[mt * 16 + v + hi * 8][wave * 16 + l16] = acc[mt][v];
  __syncthreads();
  if (threadIdx.x < 64) {
    int o = threadIdx.x;
    float s = 0.f, s2 = 0.f;
    for (int p = 0; p < 4; ++p) {
      float mx = -3.0e38f, mn = 3.0e38f;
      for (int kk = 0; kk < KNN; ++kk) {
        float v = h[p * KNN + kk][o];
        mx = fmaxf(mx, v);
        mn = fminf(mn, v);
        s += v;
        s2 += v * v;
      }
      size_t bn = (size_t)batch * NPTS + nbase + p;
      hmax[bn * 64 + o] = mx;
      hmin[bn * 64 + o] = mn;
    }
    atomicAdd(&sums[o], s);
    atomicAdd(&sumsq[o], s2);
  }
}

// ---------------------------------------------------------------------------
// Apply BN+LeakyReLU and transpose to (B, 64, N)
// ---------------------------------------------------------------------------
__global__ __launch_bounds__(256) void apply2_kernel(
    const float* __restrict__ hmax, const float* __restrict__ hmin,
    const float* __restrict__ scale, const float* __restrict__ shift,
    float* __restrict__ out) {
  int n = blockIdx.x * 256 + threadIdx.x;
  int o = blockIdx.y;
  int b = blockIdx.z;
  size_t bn = (size_t)b * NPTS + n;
  float sc = scale[o];
  float hv = (sc >= 0.f) ? hmax[bn * 64 + o] : hmin[bn * 64 + o];
  float v = sc * hv + shift[o];
  v = (v >= 0.f) ? v : NEG_SLOPE * v;
  out[((size_t)b * 64 + o) * NPTS + n] = v;
}

// ---------------------------------------------------------------------------
extern "C" void kernel_launch(void* const* d_in, const int* in_sizes, int n_in,
                              void* d_out, int out_size, void* d_ws, size_t ws_size,
                              hipStream_t stream) {
  (void)in_sizes; (void)n_in; (void)out_size; (void)ws_size;
  const float* a = (const float*)d_in[0];
  const float* b = (const float*)d_in[1];
  const float* W1 = (const float*)d_in[2];
  const float* g1 = (const float*)d_in[3];
  const float* be1 = (const float*)d_in[4];
  const float* W2 = (const float*)d_in[5];
  const float* g2 = (const float*)d_in[6];
  const float* be2 = (const float*)d_in[7];
  float* out = (float*)d_out;

  char* ws = (char*)d_ws;
  size_t off = 0;
  auto walloc = [&](size_t bytes) -> void* {
    void* p = ws + off;
    off += (bytes + 255) & ~(size_t)255;
    return p;
  };
  int* idx1 = (int*)walloc((size_t)BATCH * NPTS * KNN * 4);
  int* idx2 = (int*)walloc((size_t)BATCH * NPTS * KNN * 4);
  float* hmax = (float*)walloc((size_t)BATCH * NPTS * 64 * 4);
  float* hmin = (float*)walloc((size_t)BATCH * NPTS * 64 * 4);
  float* x1 = (float*)walloc((size_t)BATCH * NPTS * 64 * 4);
  float* xnorm2 = (float*)walloc((size_t)BATCH * NPTS * 4);
  float* stats = (float*)walloc(128 * 4);  // sums[64] + sumsq[64]
  float* sums = stats;
  float* sumsq = stats + 64;
  float* scale1 = (float*)walloc(64 * 4);
  float* shift1 = (float*)walloc(64 * 4);
  float* scale2 = (float*)walloc(64 * 4);
  float* shift2 = (float*)walloc(64 * 4);

  const float inv_cnt = 1.0f / ((float)BATCH * NPTS * KNN);

  // Block 1
  knn1_kernel<<<dim3(NPTS / 128, BATCH), 128, 0, stream>>>(a, b, idx1);
  hipMemsetAsync(stats, 0, 128 * 4, stream);
  edgeconv1_kernel<<<dim3(NPTS / 4, BATCH), 64, 0, stream>>>(a, idx1, W1, hmax,
                                                             hmin, sums, sumsq);
  bn_finalize_kernel<<<1, 64, 0, stream>>>(sums, sumsq, g1, be1, scale1, shift1,
                                           inv_cnt);
  apply1_kernel<<<BATCH * NPTS / 256, 256, 0, stream>>>(hmax, hmin, scale1,
                                                        shift1, x1, xnorm2);
  // Block 2
  knn2_kernel<<<dim3(NPTS / 64, BATCH), 128, 0, stream>>>(x1, xnorm2, idx2);
  hipMemsetAsync(stats, 0, 128 * 4, stream);
  edgeconv2_kernel<<<dim3(NPTS / 4, BATCH), 128, 0, stream>>>(x1, idx2, W2, hmax,
                                                              hmin, sums, sumsq);
  bn_finalize_kernel<<<1, 64, 0, stream>>>(sums, sumsq, g2, be2, scale2, shift2,
                                           inv_cnt);
  apply2_kernel<<<dim3(NPTS / 256, 64, BATCH), 256, 0, stream>>>(hmax, hmin,
                                                                 scale2, shift2,
                                                                 out);
}